// GPUFracturePreprocessor_274877906986
// MI455X (gfx1250) — compile-verified
//
#include <hip/hip_runtime.h>

#define TILE 32
#define HALO 2
#define IN_T   (TILE + 2*HALO)   // 36
#define IN_S   IN_T              // 36: unpadded, matches TDM contiguous rows
#define IX_T   (TILE + 2)        // 34
#define IX_S   (IX_T + 1)        // 35
#define WIMG 224
#define HIMG 224
#define PLANE (WIMG*HIMG)        // 50176
#define EPSV 1e-6f

typedef float v4f __attribute__((ext_vector_type(4)));
typedef unsigned int u32x4 __attribute__((ext_vector_type(4)));
typedef int          i32x4 __attribute__((ext_vector_type(4)));
typedef int          i32x8 __attribute__((ext_vector_type(8)));

typedef __attribute__((address_space(1))) int as1_i32;   // global
typedef __attribute__((address_space(3))) int as3_i32;   // LDS

#define HAS_ASYNC (__has_builtin(__builtin_amdgcn_global_load_async_to_lds_b32) && \
                   __has_builtin(__builtin_amdgcn_s_wait_asynccnt))
#define HAS_TDM   (__has_builtin(__builtin_amdgcn_tensor_load_to_lds) && \
                   __has_builtin(__builtin_amdgcn_s_wait_tensorcnt))

// ---------------------------------------------------------------------------
// Tile loader: 36x36 zero-padded halo tile of the structural channel into LDS.
//  * Interior-ish tiles (halo not crossing top/left image edge): one TDM
//    descriptor per block; tensor_dim0/1 = remaining image extent so hardware
//    OOB zero-fill provides the right/bottom padding.
//  * Top/left edge tiles: per-lane async global->LDS (EXEC-masked), zeros via
//    DS stores for out-of-image lanes.
// ---------------------------------------------------------------------------
__device__ __forceinline__ void tile_load(
    const float* __restrict__ plane0, int ty0, int tx0, int tid, float* s_in)
{
#if HAS_TDM && HAS_ASYNC
  if (ty0 >= HALO && tx0 >= HALO) {
    if (tid < 32) {   // one wave issues the DMA (EXEC ignored by tensor ops)
      unsigned int lds_off = (unsigned int)(uintptr_t)(as3_i32*)s_in;
      const float* gp = plane0 + (ty0 - HALO) * WIMG + (tx0 - HALO);
      unsigned long long ga = (unsigned long long)(uintptr_t)gp;
      unsigned int dim0 = (unsigned int)(WIMG - (tx0 - HALO)); // valid cols from tile start
      unsigned int dim1 = (unsigned int)(HIMG - (ty0 - HALO)); // valid rows from tile start

      // D# group0: [1:0]=count=1 | lds_addr | global_addr[56:0] | type=2
      u32x4 g0;
      g0[0] = 1u;
      g0[1] = lds_off;
      g0[2] = (unsigned int)ga;
      g0[3] = ((unsigned int)(ga >> 32) & 0x01FFFFFFu) | (2u << 30);

      // D# group1: data_size=4B(2)<<16 ; tensor_dim0/1 ; tile 36x36 ; stride0=224
      i32x8 g1;
      g1[0] = (int)(2u << 16);
      g1[1] = (int)((dim0 & 0xFFFFu) << 16);
      g1[2] = (int)((dim0 >> 16) | ((dim1 & 0xFFFFu) << 16));
      g1[3] = (int)((dim1 >> 16) | ((unsigned)IN_T << 16));  // tile_dim0 = 36
      g1[4] = (int)IN_T;                                     // tile_dim1 = 36
      g1[5] = WIMG;                                          // tensor_dim0_stride lo
      g1[6] = 0;
      g1[7] = 0;

      i32x4 g2 = (i32x4)0;
      i32x4 g3 = (i32x4)0;
#if defined(__clang_major__) && (__clang_major__ >= 23)
      i32x8 g4 = (i32x8)0;
      __builtin_amdgcn_tensor_load_to_lds(g0, g1, g2, g3, g4, 0);
#else
      __builtin_amdgcn_tensor_load_to_lds(g0, g1, g2, g3, 0);
#endif
      __builtin_amdgcn_s_wait_tensorcnt(0);
    }
  } else {
    for (int i = tid; i < IN_T * IN_T; i += 256) {
      int r = i / IN_T, c = i % IN_T;
      int gy = ty0 - HALO + r, gx = tx0 - HALO + c;
      if (gy >= 0 && gy < HIMG && gx >= 0 && gx < WIMG)
        __builtin_amdgcn_global_load_async_to_lds_b32(
            (as1_i32*)(plane0 + gy * WIMG + gx),
            (as3_i32*)&s_in[r * IN_S + c], 0, 0);
      else
        s_in[r * IN_S + c] = 0.0f;
    }
    __builtin_amdgcn_s_wait_asynccnt(0);
  }
#elif HAS_ASYNC
  for (int i = tid; i < IN_T * IN_T; i += 256) {
    int r = i / IN_T, c = i % IN_T;
    int gy = ty0 - HALO + r, gx = tx0 - HALO + c;
    if (gy >= 0 && gy < HIMG && gx >= 0 && gx < WIMG)
      __builtin_amdgcn_global_load_async_to_lds_b32(
          (as1_i32*)(plane0 + gy * WIMG + gx),
          (as3_i32*)&s_in[r * IN_S + c], 0, 0);
    else
      s_in[r * IN_S + c] = 0.0f;
  }
  __builtin_amdgcn_s_wait_asynccnt(0);
#else
  for (int i = tid; i < IN_T * IN_T; i += 256) {
    int r = i / IN_T, c = i % IN_T;
    int gy = ty0 - HALO + r, gx = tx0 - HALO + c;
    float v = 0.0f;
    if (gy >= 0 && gy < HIMG && gx >= 0 && gx < WIMG)
      v = plane0[gy * WIMG + gx];
    s_in[r * IN_S + c] = v;
  }
#endif
  __syncthreads();
}

// ---------------------------------------------------------------------------
// Shared tile computation (after s_in is loaded): Sobel ix/iy on 34x34 with
// exact chained-SAME-conv zero-padding semantics, then per-thread 2x2 quad:
// Haar fwd->gain->inv reconstruction and ridge |ixx+iyy|.
// ---------------------------------------------------------------------------
__device__ __forceinline__ void tile_compute(
    const float* __restrict__ xin, int b, int ty0, int tx0, int tid,
    float* s_in, float* s_ix, float* s_iy,
    float rec[4], float ves[4], int& qy, int& qx)
{
  const float* plane0 = xin + (size_t)b * (size_t)(3 * PLANE); // channel 0
  tile_load(plane0, ty0, tx0, tid, s_in);

  // ---- first Sobel stage on 34x34 (input-rel rows/cols 1..34) ----
  for (int i = tid; i < IX_T * IX_T; i += 256) {
    int r = i / IX_T, c = i % IX_T;
    int gy = ty0 - 1 + r, gx = tx0 - 1 + c;
    float vx = 0.0f, vy = 0.0f;
    if (gy >= 0 && gy < HIMG && gx >= 0 && gx < WIMG) {
      const float* p = s_in + (r + 1) * IN_S + (c + 1);
      float tl = p[-IN_S - 1], tc = p[-IN_S], tr = p[-IN_S + 1];
      float ml = p[-1],                       mr = p[1];
      float bl = p[ IN_S - 1], bc = p[ IN_S], br = p[ IN_S + 1];
      vx = (tr + 2.0f * mr + br) - (tl + 2.0f * ml + bl);
      vy = (bl + 2.0f * bc + br) - (tl + 2.0f * tc + tr);
    }
    s_ix[r * IX_S + c] = vx;
    s_iy[r * IX_S + c] = vy;
  }
  __syncthreads();

  qy = tid >> 4;       // 0..15
  qx = tid & 15;       // 0..15

  // ---- Haar wavelet boost for this thread's 2x2 block ----
  {
    const float* p = s_in + (HALO + 2 * qy) * IN_S + (HALO + 2 * qx);
    float a = p[0], bb = p[1], cc = p[IN_S], dd = p[IN_S + 1];
    float c0 = 0.5f * (a + bb + cc + dd) * 0.1f;
    float c1 = 0.5f * (a - bb + cc - dd) * 2.5f;
    float c2 = 0.5f * (a + bb - cc - dd) * 2.5f;
    float c3 = 0.5f * (a - bb - cc + dd) * 3.5f;
    rec[0] = 0.5f * (c0 + c1 + c2 + c3);
    rec[1] = 0.5f * (c0 - c1 + c2 - c3);
    rec[2] = 0.5f * (c0 + c1 - c2 - c3);
    rec[3] = 0.5f * (c0 - c1 - c2 + c3);
  }

  // ---- second Sobel stage (ixx+iyy) at the 4 quad pixels ----
  for (int k = 0; k < 4; ++k) {
    int oy = 2 * qy + (k >> 1), ox = 2 * qx + (k & 1);   // tile-relative
    const float* px = s_ix + (oy + 1) * IX_S + (ox + 1);
    const float* py = s_iy + (oy + 1) * IX_S + (ox + 1);
    float ixx = (px[-IX_S + 1] + 2.0f * px[1]     + px[IX_S + 1])
              - (px[-IX_S - 1] + 2.0f * px[-1]    + px[IX_S - 1]);
    float iyy = (py[ IX_S - 1] + 2.0f * py[IX_S]  + py[IX_S + 1])
              - (py[-IX_S - 1] + 2.0f * py[-IX_S] + py[-IX_S + 1]);
    ves[k] = fabsf(ixx + iyy);
  }
}

// ---------------------------------------------------------------------------
// Pass 1: per-block {recMin, recMax, vesMin, vesMax} partials (deterministic,
// no atomics).
// ---------------------------------------------------------------------------
__global__ __launch_bounds__(256) void pass1_minmax(
    const float* __restrict__ xin, float* __restrict__ partials)
{
  __shared__ float s_in[IN_T * IN_S];
  __shared__ float s_ix[IX_T * IX_S];
  __shared__ float s_iy[IX_T * IX_S];
  __shared__ v4f   s_red[256];

  int tid = threadIdx.x;
  int tx0 = blockIdx.x * TILE, ty0 = blockIdx.y * TILE, b = blockIdx.z;
  float rec[4], ves[4]; int qy, qx;
  tile_compute(xin, b, ty0, tx0, tid, s_in, s_ix, s_iy, rec, ves, qy, qx);

  v4f acc;
  acc.x = fminf(fminf(rec[0], rec[1]), fminf(rec[2], rec[3]));
  acc.y = fmaxf(fmaxf(rec[0], rec[1]), fmaxf(rec[2], rec[3]));
  acc.z = fminf(fminf(ves[0], ves[1]), fminf(ves[2], ves[3]));
  acc.w = fmaxf(fmaxf(ves[0], ves[1]), fmaxf(ves[2], ves[3]));
  s_red[tid] = acc;
  __syncthreads();

  for (int off = 128; off > 0; off >>= 1) {
    if (tid < off) {
      v4f a = s_red[tid], c = s_red[tid + off];
      a.x = fminf(a.x, c.x); a.y = fmaxf(a.y, c.y);
      a.z = fminf(a.z, c.z); a.w = fmaxf(a.w, c.w);
      s_red[tid] = a;
    }
    __syncthreads();
  }
  if (tid == 0) {
    int bl = (blockIdx.z * gridDim.y + blockIdx.y) * gridDim.x + blockIdx.x;
    ((v4f*)partials)[bl] = s_red[0];
  }
}

// ---------------------------------------------------------------------------
// Single-block fold of all partials -> {recMin, 1/(range+eps), vesMin, inv}.
// ---------------------------------------------------------------------------
__global__ __launch_bounds__(256) void reduce_final(
    const float* __restrict__ partials, float* __restrict__ finals, int n)
{
  __shared__ v4f s_red[256];
  int tid = threadIdx.x;
  v4f acc;
  acc.x =  __builtin_inff(); acc.y = -__builtin_inff();
  acc.z =  __builtin_inff(); acc.w = -__builtin_inff();
  const v4f* p4 = (const v4f*)partials;
  for (int i = tid; i < n; i += 256) {
    v4f p = p4[i];
    acc.x = fminf(acc.x, p.x); acc.y = fmaxf(acc.y, p.y);
    acc.z = fminf(acc.z, p.z); acc.w = fmaxf(acc.w, p.w);
  }
  s_red[tid] = acc;
  __syncthreads();
  for (int off = 128; off > 0; off >>= 1) {
    if (tid < off) {
      v4f a = s_red[tid], c = s_red[tid + off];
      a.x = fminf(a.x, c.x); a.y = fmaxf(a.y, c.y);
      a.z = fminf(a.z, c.z); a.w = fmaxf(a.w, c.w);
      s_red[tid] = a;
    }
    __syncthreads();
  }
  if (tid == 0) {
    v4f r = s_red[0];
    finals[0] = r.x;
    finals[1] = 1.0f / (r.y - r.x + EPSV);
    finals[2] = r.z;
    finals[3] = 1.0f / (r.w - r.z + EPSV);
  }
}

// ---------------------------------------------------------------------------
// Pass 2: recompute tile (input is L2-hot), normalize, 128-bit non-temporal
// stores of all three output channels.
// ---------------------------------------------------------------------------
__global__ __launch_bounds__(256) void pass2_write(
    const float* __restrict__ xin,
    const float* __restrict__ meanv, const float* __restrict__ stdv,
    const float* __restrict__ finals, float* __restrict__ out)
{
  __shared__ float s_in[IN_T * IN_S];
  __shared__ float s_ix[IX_T * IX_S];
  __shared__ float s_iy[IX_T * IX_S];
  __shared__ float s_rec[TILE * (TILE + 1)];
  __shared__ float s_ves[TILE * (TILE + 1)];

  int tid = threadIdx.x;
  int tx0 = blockIdx.x * TILE, ty0 = blockIdx.y * TILE, b = blockIdx.z;
  float rec[4], ves[4]; int qy, qx;
  tile_compute(xin, b, ty0, tx0, tid, s_in, s_ix, s_iy, rec, ves, qy, qx);

  for (int k = 0; k < 4; ++k) {
    int oy = 2 * qy + (k >> 1), ox = 2 * qx + (k & 1);
    s_rec[oy * (TILE + 1) + ox] = rec[k];
    s_ves[oy * (TILE + 1) + ox] = ves[k];
  }
  __syncthreads();

  float recMin = finals[0], recInv = finals[1];
  float vesMin = finals[2], vesInv = finals[3];
  float m0 = meanv[0], m1 = meanv[1], m2 = meanv[2];
  float i0 = 1.0f / stdv[0], i1 = 1.0f / stdv[1], i2 = 1.0f / stdv[2];

  int row  = tid >> 3;          // 0..31
  int col  = (tid & 7) * 4;     // 0,4,...,28
  size_t base = (size_t)b * (size_t)(3 * PLANE)
              + (size_t)(ty0 + row) * WIMG + (size_t)(tx0 + col);

  v4f o0, o1, o2;
  for (int j = 0; j < 4; ++j) {
    float sx = s_in [(row + HALO) * IN_S + (col + j + HALO)];
    float fr = (s_rec[row * (TILE + 1) + col + j] - recMin) * recInv;
    float rd = (s_ves[row * (TILE + 1) + col + j] - vesMin) * vesInv;
    o0[j] = (sx - m0) * i0;
    o1[j] = (fr - m1) * i1;
    o2[j] = (rd - m2) * i2;
  }
  __builtin_nontemporal_store(o0, (v4f*)(out + base));
  __builtin_nontemporal_store(o1, (v4f*)(out + base + PLANE));
  __builtin_nontemporal_store(o2, (v4f*)(out + base + 2 * PLANE));
}

// ---------------------------------------------------------------------------
extern "C" void kernel_launch(void* const* d_in, const int* in_sizes, int n_in,
                              void* d_out, int out_size, void* d_ws, size_t ws_size,
                              hipStream_t stream)
{
  const float* x     = (const float*)d_in[0];
  const float* meanv = (const float*)d_in[2];
  const float* stdv  = (const float*)d_in[3];
  float* out = (float*)d_out;

  int batches = in_sizes[0] / (3 * PLANE);   // 256 for the reference shapes

  float* finals   = (float*)d_ws;            // 4 floats
  float* partials = (float*)d_ws + 16;       // 64B-aligned v4f array

  dim3 grid(WIMG / TILE, HIMG / TILE, batches);  // 7 x 7 x B
  dim3 block(256);

  pass1_minmax<<<grid, block, 0, stream>>>(x, partials);
  reduce_final<<<1, 256, 0, stream>>>(partials, finals,
                                      (WIMG / TILE) * (HIMG / TILE) * batches);
  pass2_write<<<grid, block, 0, stream>>>(x, meanv, stdv, finals, out);
}